// Yolov2Loss_69286412419448
// MI455X (gfx1250) — compile-verified
//
#include <hip/hip_runtime.h>
#include <stdint.h>

// ---------------- problem constants (from reference) ----------------
#define IGNORE_T    0.6f
#define L_NOOBJ     0.5f
#define L_PRIOR     0.01f
#define L_COORD     5.0f
#define L_OBJ       1.0f
#define L_CLS       1.0f
#define PRIOR_EPOCHS 12
#define EPS_IOU     1e-8f

#define BS    16
#define GRD   19
#define NA    5
#define NC    20
#define ROW   (5 + NC)          // 25 floats per cell
#define NBOX  (GRD * GRD * NA)  // 1805 boxes per batch image
#define NCELL (BS * NBOX)       // 28880 cells total

// ---------------- CDNA5 async global->LDS path ----------------------
#if defined(__has_builtin)
# if __has_builtin(__builtin_amdgcn_global_load_async_to_lds_b128)
#  define HAVE_ASYNC_B128 1
# endif
# if __has_builtin(__builtin_amdgcn_s_wait_asynccnt)
#  define HAVE_WAIT_ASYNC 1
# endif
#endif

// builtin prototype (from hipcc diagnostic): v4i addrspace(1)*, v4i addrspace(3)*, imm, imm
typedef int v4i_b128 __attribute__((__vector_size__(16)));
typedef __attribute__((address_space(1))) v4i_b128* g_v4i_ptr;
typedef __attribute__((address_space(3))) v4i_b128* l_v4i_ptr;

__device__ __forceinline__ void stage16_async(const float4* gsrc, float4* ldst) {
#if defined(HAVE_ASYNC_B128)
  // ASYNCcnt-tracked DMA of one 16B gt box into LDS (no VGPR data path).
  // AS3 pointers are 32-bit; low 32 bits of a flat LDS address are the LDS offset.
  __builtin_amdgcn_global_load_async_to_lds_b128(
      (g_v4i_ptr)(uintptr_t)gsrc,
      (l_v4i_ptr)(uint32_t)(uintptr_t)ldst,
      0, 0);
#else
  *ldst = *gsrc;
#endif
}

__device__ __forceinline__ void wait_async_zero() {
#if defined(HAVE_WAIT_ASYNC)
  __builtin_amdgcn_s_wait_asynccnt(0);
#elif defined(HAVE_ASYNC_B128)
  asm volatile("s_wait_asynccnt 0" ::: "memory");
#endif
}

// ---------------- helpers -------------------------------------------
__device__ __forceinline__ float sigmoidf(float x) {
  return 1.0f / (1.0f + expf(-x));
}

// wave32 block reduction (gfx1250 is wave32-only)
__device__ __forceinline__ float block_sum(float v) {
  #pragma unroll
  for (int off = 16; off > 0; off >>= 1) v += __shfl_down(v, off, 32);
  __shared__ float smem[32];
  const int lane = threadIdx.x & 31;
  const int wid  = threadIdx.x >> 5;
  if (lane == 0) smem[wid] = v;
  __syncthreads();
  const int nw = (blockDim.x + 31) >> 5;
  v = (threadIdx.x < (unsigned)nw) ? smem[threadIdx.x] : 0.0f;
  if (wid == 0) {
    #pragma unroll
    for (int off = 16; off > 0; off >>= 1) v += __shfl_down(v, off, 32);
  }
  return v;  // valid on thread 0
}

__device__ __forceinline__ float iou_xyxy(float ax1, float ay1, float ax2, float ay2,
                                          float bx1, float by1, float bx2, float by2) {
  float ix1 = fmaxf(ax1, bx1), iy1 = fmaxf(ay1, by1);
  float ix2 = fminf(ax2, bx2), iy2 = fminf(ay2, by2);
  float iw = fmaxf(ix2 - ix1, 0.0f), ih = fmaxf(iy2 - iy1, 0.0f);
  float inter = iw * ih;
  float aa = fmaxf(ax2 - ax1, 0.0f) * fmaxf(ay2 - ay1, 0.0f);
  float ab = fmaxf(bx2 - bx1, 0.0f) * fmaxf(by2 - by1, 0.0f);
  return inter / (aa + ab - inter + EPS_IOU);
}

// ---------------- kernel 0: init ------------------------------------
__global__ void k_init(float* __restrict__ out, int* __restrict__ counts) {
  if (threadIdx.x == 0) out[0] = 0.0f;
  if (threadIdx.x < BS) counts[threadIdx.x] = 0;
}

// ---------------- kernel 1: elementwise losses + compaction ---------
__global__ __launch_bounds__(256)
void k_elemwise(const float* __restrict__ pred,
                const float* __restrict__ gt,
                const float* __restrict__ anchors,
                const int*   __restrict__ epoch_p,
                float* __restrict__ out,
                int*   __restrict__ counts,
                float4* __restrict__ gtc,       // [BS*NBOX] compacted positive gt boxes
                float4* __restrict__ pbox_arr,  // [NCELL] decoded pred boxes
                float2* __restrict__ aux_arr) { // [NCELL] (pconf, pos)
  const int e = blockIdx.x * blockDim.x + threadIdx.x;
  float loss = 0.0f;
  if (e < NCELL) {
    int a  = e % NA;
    int t  = e / NA;
    int i2 = t % GRD;  t /= GRD;     // fast grid axis -> added to x
    int i1 = t % GRD;                // slow grid axis -> added to y
    int b  = t / GRD;

    const float* p = pred + (size_t)e * ROW;
    const float* g = gt   + (size_t)e * ROW;
    __builtin_prefetch(g, 0, 1);     // global_prefetch_b8

    const float aw = anchors[2 * a], ah = anchors[2 * a + 1];
    const float px = p[0], py = p[1], pw = p[2], ph = p[3], po = p[4];
    const float gx = g[0], gy = g[1], gw = g[2], gh = g[3], go = g[4];

    const float sx = sigmoidf(px), sy = sigmoidf(py);
    const float pconf = sigmoidf(po);
    const float Gf = (float)GRD;
    const float phw = 0.5f * expf(pw) * aw * Gf, phh = 0.5f * expf(ph) * ah * Gf;
    const float ghw = 0.5f * expf(gw) * aw * Gf, ghh = 0.5f * expf(gh) * ah * Gf;

    const float cxp = sx + (float)i2, cyp = sy + (float)i1;
    const float cxg = gx + (float)i2, cyg = gy + (float)i1;
    const float px1 = cxp - phw, py1 = cyp - phh, px2 = cxp + phw, py2 = cyp + phh;
    const float gx1 = cxg - ghw, gy1 = cyg - ghh, gx2 = cxg + ghw, gy2 = cyg + ghh;

    // prior loss (all cells, epoch-gated)
    if (*epoch_p < PRIOR_EPOCHS)
      loss += L_PRIOR * (px * px + py * py + pw * pw + ph * ph);

    const bool pos = (go > 0.5f);
    if (pos) {
      // coord loss: (gt_xywh - [sig(x),sig(y),w,h])^2
      float d0 = gx - sx, d1 = gy - sy, d2 = gw - pw, d3 = gh - ph;
      loss += L_COORD * (d0 * d0 + d1 * d1 + d2 * d2 + d3 * d3);
      // conf loss: (diag IoU - pconf)^2
      float ci = iou_xyxy(px1, py1, px2, py2, gx1, gy1, gx2, gy2);
      float dc = ci - pconf;
      loss += L_OBJ * dc * dc;
      // class loss: (gcls - softmax(pcls))^2
      float m = p[5];
      #pragma unroll
      for (int k = 1; k < NC; ++k) m = fmaxf(m, p[5 + k]);
      float ex[NC], s = 0.0f;
      #pragma unroll
      for (int k = 0; k < NC; ++k) { ex[k] = expf(p[5 + k] - m); s += ex[k]; }
      const float inv = 1.0f / s;
      float cl = 0.0f;
      #pragma unroll
      for (int k = 0; k < NC; ++k) { float d = g[5 + k] - ex[k] * inv; cl += d * d; }
      loss += L_CLS * cl;
      // compact this positive gt box for the O(N^2) pass (max is order-invariant)
      int slot = atomicAdd(&counts[b], 1);
      gtc[b * NBOX + slot] = make_float4(gx1, gy1, gx2, gy2);
    }
    pbox_arr[e] = make_float4(px1, py1, px2, py2);
    aux_arr[e]  = make_float2(pconf, pos ? 1.0f : 0.0f);
  }
  float bsum = block_sum(loss);
  if (threadIdx.x == 0) atomicAdd(out, bsum * (1.0f / (float)BS));
}

// ---------------- kernel 2: all-pairs max-IoU + noobj loss ----------
__global__ __launch_bounds__(256)
void k_maxiou_noobj(const int*    __restrict__ counts,
                    const float4* __restrict__ gtc,
                    const float4* __restrict__ pbox_arr,
                    const float2* __restrict__ aux_arr,
                    float* __restrict__ out) {
  __shared__ float4 sbox[NBOX];   // <= 28.9 KB (WGP has 320 KB LDS)
  __shared__ float  sarea[NBOX];  // 7.2 KB
  const int b    = blockIdx.y;
  const int nPos = counts[b];
  const float4* src = gtc + (size_t)b * NBOX;

  // async DMA the compacted gt-box set into LDS (ASYNCcnt path)
  for (int j = threadIdx.x; j < nPos; j += blockDim.x)
    stage16_async(src + j, &sbox[j]);
  wait_async_zero();
  __syncthreads();
  for (int j = threadIdx.x; j < nPos; j += blockDim.x) {
    float4 gb = sbox[j];
    sarea[j] = fmaxf(gb.z - gb.x, 0.0f) * fmaxf(gb.w - gb.y, 0.0f);
  }
  __syncthreads();

  const int i = blockIdx.x * blockDim.x + threadIdx.x;  // pred index within batch
  float loss = 0.0f;
  if (i < NBOX) {
    const float4 pb = pbox_arr[(size_t)b * NBOX + i];
    const float2 ax = aux_arr[(size_t)b * NBOX + i];
    const float areaP = fmaxf(pb.z - pb.x, 0.0f) * fmaxf(pb.w - pb.y, 0.0f);
    float m = 0.0f;  // IoU >= 0, masked terms contribute 0
    for (int j = 0; j < nPos; ++j) {     // uniform j -> LDS broadcast, no conflicts
      float4 gb = sbox[j];
      float ix1 = fmaxf(pb.x, gb.x), iy1 = fmaxf(pb.y, gb.y);
      float ix2 = fminf(pb.z, gb.z), iy2 = fminf(pb.w, gb.w);
      float iw = fmaxf(ix2 - ix1, 0.0f), ih = fmaxf(iy2 - iy1, 0.0f);
      float inter = iw * ih;
      float iou = inter / (areaP + sarea[j] - inter + EPS_IOU);
      m = fmaxf(m, iou);
    }
    if (ax.y < 0.5f && m < IGNORE_T) loss = L_NOOBJ * ax.x * ax.x;
  }
  float bsum = block_sum(loss);
  if (threadIdx.x == 0) atomicAdd(out, bsum * (1.0f / (float)BS));
}

// ---------------- launcher ------------------------------------------
extern "C" void kernel_launch(void* const* d_in, const int* in_sizes, int n_in,
                              void* d_out, int out_size, void* d_ws, size_t ws_size,
                              hipStream_t stream) {
  (void)in_sizes; (void)n_in; (void)out_size; (void)ws_size;
  const float* pred    = (const float*)d_in[0];
  const float* gt      = (const float*)d_in[1];
  const float* anchors = (const float*)d_in[2];
  const int*   epoch   = (const int*)d_in[3];
  float* out = (float*)d_out;

  // workspace layout (bytes): counts[16] @0 (pad 256) | gtc 462080 | pbox 462080 | aux 231040
  char* ws = (char*)d_ws;
  int*    counts = (int*)ws;
  float4* gtc    = (float4*)(ws + 256);
  float4* pbox   = (float4*)(ws + 256 + (size_t)NBOX * BS * sizeof(float4));
  float2* aux    = (float2*)(ws + 256 + 2 * (size_t)NBOX * BS * sizeof(float4));

  k_init<<<1, 32, 0, stream>>>(out, counts);
  k_elemwise<<<(NCELL + 255) / 256, 256, 0, stream>>>(pred, gt, anchors, epoch,
                                                      out, counts, gtc, pbox, aux);
  dim3 g2((NBOX + 255) / 256, BS);
  k_maxiou_noobj<<<g2, 256, 0, stream>>>(counts, gtc, pbox, aux, out);
}